// PilotAEQ_SP_26001732010263
// MI455X (gfx1250) — compile-verified
//
#include <hip/hip_runtime.h>
#include <hip/hip_bf16.h>

// PilotAEQ: pilot-aided LMS equalizer.
//   Phase 1 (i = 0..16376): exact serial LMS recurrence -> single wave32,
//     one complex tap per lane, ds-swizzle butterfly reduction per step,
//     global_prefetch_b8 to pre-warm the sliding window.
//   Phase 2 (i = 16377..65519): frozen-tap 31-tap complex FIR at stride 2,
//     expressed as f32 WMMA 16x16x4 chains (one 16-output tile per wave).
//
// Inputs (complex64 viewed as float2):
//   d_in[0] y        [131072]
//   d_in[1] pilot    [32768]
//   d_in[2] pilot_up [65536]
//   d_in[3] taps0    [31]
// Output: complex64 [65620] = 100 zeros + 65520 equalized symbols.

#define SPS        2
#define LTAPS      31
#define NUM_OUT    65520          // (131072 - 31) / 2
#define I_FREEZE   16377          // first i with (2i + 14) >= 32768  (taps frozen)
#define LR2        (2.0e-3f)      // 2 * LR
#define TILES      3072           // ceil((NUM_OUT - I_FREEZE) / 16)
#define WAVES_PB   8

typedef __attribute__((ext_vector_type(2))) float v2f;
typedef __attribute__((ext_vector_type(8))) float v8f;

// ---------------------------------------------------------------------------
// Phase 1: serial LMS, one wave32. Lane m (< 31) owns tap[m].
// out_i = sum_m y[2i + 30 - m] * tap[m]      (flip folded into the y index)
// tap[m] += 2*LR * e_i * conj(pilot_up[2i + 30 - m])
// ---------------------------------------------------------------------------
__global__ __launch_bounds__(32) void aeq_lms_seq(
    const float2* __restrict__ y,
    const float2* __restrict__ pilot,
    const float2* __restrict__ pu,
    const float2* __restrict__ taps0,
    float2* __restrict__ out,          // d_out viewed as float2[65620]
    float2* __restrict__ taps_ws)      // d_ws: final taps [31]
{
    const int  lane  = threadIdx.x;          // 0..31
    const bool isTap = lane < LTAPS;

    float2 t = isTap ? taps0[lane] : make_float2(0.f, 0.f);

    // 100 leading zeros of the output
    for (int z = lane; z < 100; z += 32) out[z] = make_float2(0.f, 0.f);

    const int ridx = isTap ? (30 - lane) : 0;   // window offset for this lane

    for (int i = 0; i < I_FREEZE; ++i) {
        const int k0 = 2 * i;

        // keep the sliding window warm in WGP$/L2 (global_prefetch_b8)
        if ((i & 63) == 0) {
            __builtin_prefetch(y  + k0 + 512, 0, 1);
            __builtin_prefetch(pu + k0 + 512, 0, 1);
        }

        const float2 w = y[k0 + ridx];
        float pr = isTap ? (w.x * t.x - w.y * t.y) : 0.f;
        float pi = isTap ? (w.x * t.y + w.y * t.x) : 0.f;

        // wave32 butterfly: all lanes end with the full complex sum
        #pragma unroll
        for (int s = 16; s >= 1; s >>= 1) {
            pr += __shfl_xor(pr, s, 32);
            pi += __shfl_xor(pi, s, 32);
        }

        if (lane == 0) out[100 + i] = make_float2(pr, pi);

        const float2 p  = pilot[7 + i];        // 7+i <= 16383 < 32768: no clamp
        const float  er = p.x - pr;
        const float  ei = p.y - pi;

        // u[m] = e * conj(pu[2i + 30 - m]);  tap[m] += 2*LR * u[m]
        const float2 q = pu[k0 + ridx];
        if (isTap) {
            t.x += LR2 * (er * q.x + ei * q.y);
            t.y += LR2 * (ei * q.x - er * q.y);
        }
    }

    if (isTap) taps_ws[lane] = t;
}

// ---------------------------------------------------------------------------
// Phase 2: frozen-tap FIR via V_WMMA_F32_16X16X4_F32.
// Real-valued view: out_re(i) = sum_{u<62} yf[4i+u] * cr[u]
//                   out_im(i) = sum_{u<62} yf[4i+u] * ci[u]
// cr[2t] =  Re taps[30-t], cr[2t+1] = -Im taps[30-t]
// ci[2t] =  Im taps[30-t], ci[2t+1] =  Re taps[30-t]
// Tile: 16 outputs (rows), B columns 0/1 = cr/ci, K padded to 64 -> 16 WMMAs.
//
// f32 layouts (ISA 7.12.2): A 16x4: lane L -> M=L&15, regs K={0,1} (L<16) or
// {2,3} (L>=16). B 4x16 mirrors (N in lanes, same K pairing). D: reg v,
// lane L -> [v + (L<16?0:8)][L&15].
// ---------------------------------------------------------------------------
__global__ __launch_bounds__(256) void aeq_fir_wmma(
    const float* __restrict__ yf,          // y as flat float[262144]
    const float2* __restrict__ taps_ws,
    float* __restrict__ outf)              // d_out as flat float[131240]
{
    __shared__ float cr[64];
    __shared__ float ci[64];

    const int tid = threadIdx.x;
    if (tid < 64) {
        float vr = 0.f, vi = 0.f;
        if (tid < 2 * LTAPS) {
            const float2 tap = taps_ws[30 - (tid >> 1)];
            if ((tid & 1) == 0) { vr = tap.x;  vi = tap.y; }
            else                { vr = -tap.y; vi = tap.x; }
        }
        cr[tid] = vr;
        ci[tid] = vi;
    }
    __syncthreads();

    const int lane = tid & 31;
    const int wave = tid >> 5;
    const int tile = blockIdx.x * WAVES_PB + wave;
    if (tile >= TILES) return;                       // wave-uniform

    const int  i0   = I_FREEZE + tile * 16;
    const int  N    = lane & 15;                     // A: row M == N; B/D: col N
    const int  kA   = (lane < 16) ? 0 : 2;
    const int  irow = min(i0 + N, NUM_OUT - 1);      // clamp loads, mask stores
    const long rbase = 4L * irow;

    const float sel0 = (N == 0) ? 1.f : 0.f;         // column 0 carries cr
    const float sel1 = (N == 1) ? 1.f : 0.f;         // column 1 carries ci

    v8f acc = {};
    #pragma unroll
    for (int c = 0; c < 16; ++c) {
        const int u = 4 * c + kA;
        v2f a, b;
        a.x = yf[rbase + u];
        a.y = yf[rbase + u + 1];
        b.x = cr[u]     * sel0 + ci[u]     * sel1;   // LDS broadcast reads
        b.y = cr[u + 1] * sel0 + ci[u + 1] * sel1;
        // (neg_a, A, neg_b, B, c_mod, C, reuse_a, reuse_b)
        acc = __builtin_amdgcn_wmma_f32_16x16x4_f32(
            false, a, false, b, (short)0, acc, false, false);
    }

    // D columns 0 (re) and 1 (im) live in lanes 0,1,16,17
    if (N < 2) {
        const int roff = (lane < 16) ? 0 : 8;
        #pragma unroll
        for (int v = 0; v < 8; ++v) {
            const int i = i0 + v + roff;
            if (i < NUM_OUT) outf[(100 + i) * 2 + N] = acc[v];
        }
    }
}

extern "C" void kernel_launch(void* const* d_in, const int* in_sizes, int n_in,
                              void* d_out, int out_size, void* d_ws, size_t ws_size,
                              hipStream_t stream) {
    const float2* y     = (const float2*)d_in[0];
    const float2* pilot = (const float2*)d_in[1];
    const float2* pu    = (const float2*)d_in[2];
    const float2* taps0 = (const float2*)d_in[3];
    float2* taps_ws = (float2*)d_ws;

    // Phase 1: serial recurrence (single wave), writes out[0 .. 100+I_FREEZE)
    aeq_lms_seq<<<1, 32, 0, stream>>>(y, pilot, pu, taps0,
                                      (float2*)d_out, taps_ws);

    // Phase 2: WMMA FIR for the frozen-tap tail (stream-ordered after phase 1)
    aeq_fir_wmma<<<TILES / WAVES_PB, 256, 0, stream>>>(
        (const float*)d_in[0], taps_ws, (float*)d_out);
}